// ConvCapsules2d_3212635538051
// MI455X (gfx1250) — compile-verified
//
#include <hip/hip_runtime.h>

// ConvCapsules2d on MI455X (gfx1250).
//
// Reference is a broadcast product (no contraction):
//   V[n,i,c,d,f,g,k,l] = s[n,i,f,g,k,l] * W[i,c,d,k,l]
//   s[n,i,f,g,k,l]     = sum_p poses[n,i,p, 2f+k, 2g+l]
// Output = 84.93M f32 = 339.7 MB -> store-bandwidth bound (~14.6 us @ 23.3 TB/s).
// Inputs are tiny (7 MB total) and L2-resident; recomputed per block.
//
// WMMA mapping: per (n,i,c) the contiguous 16x324 output block is
//   M = W_row(16x9) x X(9x324), X[kk,col] = (col%9==kk) ? s[col] : 0
// done with three chained V_WMMA_F32_16X16X4_F32 (K = 9 padded to 12), staged
// through LDS so global writes remain fully-coalesced 128-bit stores.
// Output drain uses non-temporal stores: the 340 MB write-once stream must not
// thrash the 192 MB L2 that holds the reusable poses/W inputs.

typedef __attribute__((ext_vector_type(2))) float v2f;
typedef __attribute__((ext_vector_type(4))) float v4f;
typedef __attribute__((ext_vector_type(8))) float v8f;

#define NI      512          // N*B_IN = 16*32
#define P_DIM   16
#define HH      14
#define HW      (HH*HH)      // 196
#define C_OUT   32
#define D_OUT   16
#define KLN     9            // K*K
#define COLS    324          // F*F*K*K, contiguous per (n,i,c,d)
#define NTILES  21           // ceil(324/16)
#define CPB     8            // c-values per block
#define BLK_C   (16*COLS)    // 5184 floats per c
#define THREADS 256

__global__ __launch_bounds__(THREADS)
void capsconv_wmma_kernel(const float* __restrict__ poses,
                          const float* __restrict__ W,
                          float* __restrict__ out) {
    __shared__ float sbuf[COLS];                 // s[n,i,fg*9+kl]
    __shared__ float wbuf[CPB * D_OUT * KLN];    // W[i, c0+cc, d, kl]
    __shared__ float obuf[BLK_C];                // one c worth of output (d, col)

    const int tid  = threadIdx.x;
    const int lane = tid & 31;
    const int wave = tid >> 5;
    const int h    = lane >> 4;     // lane half (WMMA K/M striping)
    const int j    = lane & 15;     // column within tile / row d for A

    const int ni = blockIdx.x >> 2;       // flat (n,i) in [0,512)
    const int cq = blockIdx.x & 3;        // which group of 8 c
    const int i  = ni & 31;
    const int c0 = cq * CPB;

    // ---- Phase 1: sbuf[col] = sum_p poses[ni, p, 2f+k, 2g+l] -----------------
    const float* pp = poses + (size_t)ni * (P_DIM * HW);
    for (int idx = tid; idx < COLS; idx += THREADS) {
        int fg = idx / 9, kl = idx - fg * 9;
        int f = fg / 6,  g = fg - f * 6;
        int k = kl / 3,  l = kl - k * 3;
        int off = (2 * f + k) * HH + (2 * g + l);
        float s = 0.f;
#pragma unroll
        for (int p = 0; p < P_DIM; ++p) s += pp[p * HW + off];
        sbuf[idx] = s;
    }
    // ---- Phase 2: stage 8 rows of W ------------------------------------------
    const float* wp = W + (size_t)(i * C_OUT + c0) * (D_OUT * KLN);
    for (int idx = tid; idx < CPB * D_OUT * KLN; idx += THREADS) wbuf[idx] = wp[idx];
    __syncthreads();

    for (int cc = 0; cc < CPB; ++cc) {
        // A-matrix (16x4 f32): VGPR0 holds K=4kc+{0,2} across lane halves,
        // VGPR1 holds K=4kc+{1,3}. Rows m = j map to d. Zero-pad K>=9.
        v2f a[3];
#pragma unroll
        for (int kc = 0; kc < 3; ++kc) {
            int kx = 4 * kc + 2 * h;           // K index of component .x
            int wb = cc * (D_OUT * KLN) + j * KLN;
            a[kc].x = (kx     < KLN) ? wbuf[wb + kx]     : 0.f;
            a[kc].y = (kx + 1 < KLN) ? wbuf[wb + kx + 1] : 0.f;
        }

        for (int t = wave; t < NTILES; t += 8) {
            int col  = t * 16 + j;
            bool inb = col < COLS;
            float sv = inb ? sbuf[inb ? col : 0] : 0.f;
            int  klc = col % 9;                 // which K-row of X is nonzero
            v8f acc = {};
#pragma unroll
            for (int kc = 0; kc < 3; ++kc) {
                int kx = 4 * kc + 2 * h;        // B layout mirrors A striping
                v2f b;
                b.x = (inb && klc == kx    ) ? sv : 0.f;
                b.y = (inb && klc == kx + 1) ? sv : 0.f;
                acc = __builtin_amdgcn_wmma_f32_16x16x4_f32(
                          false, a[kc], false, b, (short)0, acc, false, false);
            }
            // D layout: VGPR r -> row r (lanes 0-15) / r+8 (lanes 16-31), col = lane%16
            if (inb) {
#pragma unroll
                for (int r = 0; r < 8; ++r) {
                    int d = r + 8 * h;
                    obuf[d * COLS + col] = acc[r];
                }
            }
        }
        __syncthreads();

        // ---- Drain: coalesced 128-bit NON-TEMPORAL stores of the block -------
        const size_t gbase = (size_t)(ni * C_OUT + c0 + cc) * BLK_C;
        v4f* o4 = (v4f*)(out + gbase);
        const v4f* l4 = (const v4f*)obuf;
        for (int idx = tid; idx < BLK_C / 4; idx += THREADS)
            __builtin_nontemporal_store(l4[idx], o4 + idx);
        __syncthreads();
    }
}

extern "C" void kernel_launch(void* const* d_in, const int* in_sizes, int n_in,
                              void* d_out, int out_size, void* d_ws, size_t ws_size,
                              hipStream_t stream) {
    const float* poses = (const float*)d_in[0];
    const float* W     = (const float*)d_in[1];
    float* out         = (float*)d_out;
    (void)in_sizes; (void)n_in; (void)out_size; (void)d_ws; (void)ws_size;

    capsconv_wmma_kernel<<<dim3(NI * 4), dim3(THREADS), 0, stream>>>(poses, W, out);
}